// GPUHungarianMatcher_50276887167549
// MI455X (gfx1250) — compile-verified
//
#include <hip/hip_runtime.h>
#include <hip/hip_bf16.h>
#include <math.h>
#include <float.h>
#include <limits.h>

// GPUHungarianMatcher for MI455X (gfx1250, wave32).
// One workgroup per batch; whole JV solver state lives in LDS (320KB/WGP).
// Inputs staged into LDS with gfx1250 async global->LDS DMA (ASYNCcnt path).
// Costs computed in fp32 (as jax does), solver arithmetic in fp64 (as the
// reference's numpy float64 solve does).

#define B_  8
#define Q_  2048     // m columns of transposed problem
#define T_  256      // n rows of transposed problem
#define NTHREADS 1024

// ---------------- LDS pool layout (bytes) ----------------
#define OFF_PROB 0                       // Q_ floats  (softmax probs; logits temp)
#define OFF_OPX  (OFF_PROB + Q_*4)       // Q_ floats
#define OFF_OPY  (OFF_OPX  + Q_*4)       // Q_ floats
#define OFF_TPX  (OFF_OPY  + Q_*4)       // T_ floats
#define OFF_TPY  (OFF_TPX  + T_*4)       // T_ floats
#define OFF_ANS  (OFF_TPY  + T_*4)       // T_ ints
#define OFF_RVAL (OFF_ANS  + T_*4)       // 32 doubles (per-wave reduce)
#define OFF_RIDX (OFF_RVAL + 32*8)       // 32 ints
#define OFF_SC   (OFF_RIDX + 32*4)       // scalars (64B, 8B aligned)
#define OFF_OVL  (OFF_SC   + 64)         // overlap region: staging OR solver arrays
#define OFF_MINV (OFF_OVL)               // (Q_+1) doubles -> 16392, pad 16400
#define OFF_V    (OFF_MINV + 16400)      // (Q_+1) doubles
#define OFF_WAY  (OFF_V    + 16400)      // (Q_+1) ints
#define OFF_P    (OFF_WAY  + 8208)       // (Q_+1) ints
#define OFF_U    (OFF_P    + 8208)       // (T_+1) doubles -> 2056, pad 2064
#define OFF_USED (OFF_U    + 2064)       // (Q_+1) bytes -> 2049, pad 2064
#define POOLSZ   (OFF_USED + 2064)       // = 81440 B (~80KB; gfx1250 LDS is 320KB/WG)
// staging (24576 + 3072 = 27648 B) aliases OFF_OVL.. and is consumed before
// the solver arrays are first written.

struct Scal { double delta; float f0; int j0; int j1; int done; int pad[2]; };

typedef int v4i __attribute__((ext_vector_type(4)));
#define AS1 __attribute__((address_space(1)))
#define AS3 __attribute__((address_space(3)))

// ---- gfx1250 async global->LDS 16B copy (ASYNCcnt) ----
__device__ __forceinline__ void async_copy16(const void* g, void* l) {
#if __has_builtin(__builtin_amdgcn_global_load_async_to_lds_b128)
  __builtin_amdgcn_global_load_async_to_lds_b128(
      (AS1 v4i*)g, (AS3 v4i*)l, 0, 0);
#else
  asm volatile("global_load_async_to_lds_b128 %0, %1, off"
               :: "v"((AS3 void*)l), "v"(g) : "memory");
#endif
}

__device__ __forceinline__ void wait_async0() {
#if __has_builtin(__builtin_amdgcn_s_wait_asynccnt)
  __builtin_amdgcn_s_wait_asynccnt(0);
#else
  asm volatile("s_wait_asynccnt 0" ::: "memory");
#endif
}

__device__ __forceinline__ void wave_min_pair(double& val, int& idx) {
#pragma unroll
  for (int off = 16; off > 0; off >>= 1) {
    double ov = __shfl_xor(val, off, 32);
    int    oi = __shfl_xor(idx, off, 32);
    if (ov < val || (ov == val && oi < idx)) { val = ov; idx = oi; }
  }
}

extern "C" __global__ __launch_bounds__(NTHREADS)
void hungarian_kernel(const float* __restrict__ outputs,   // (B, Q, 3)
                      const float* __restrict__ targets,   // (B, T, 3)
                      int* __restrict__ row_out,           // (B, T)
                      int* __restrict__ col_out)           // (B, T)
{
  __shared__ __align__(16) unsigned char pool[POOLSZ];

  const int tid  = threadIdx.x;
  const int lane = tid & 31;
  const int wid  = tid >> 5;
  const int b    = blockIdx.x;

  float*  prob  = (float*)(pool + OFF_PROB);
  float*  opx   = (float*)(pool + OFF_OPX);
  float*  opy   = (float*)(pool + OFF_OPY);
  float*  tpx   = (float*)(pool + OFF_TPX);
  float*  tpy   = (float*)(pool + OFF_TPY);
  int*    ansA  = (int*)  (pool + OFF_ANS);
  double* rvalD = (double*)(pool + OFF_RVAL);
  int*    ridx  = (int*)  (pool + OFF_RIDX);
  Scal*   sc    = (Scal*) (pool + OFF_SC);
  double* minv  = (double*)(pool + OFF_MINV);
  double* v     = (double*)(pool + OFF_V);
  int*    way   = (int*)  (pool + OFF_WAY);
  int*    p     = (int*)  (pool + OFF_P);
  double* u     = (double*)(pool + OFF_U);
  unsigned char* used = (unsigned char*)(pool + OFF_USED);

  // ---------- Phase A: async-DMA stage inputs into LDS ----------
  float* stage  = (float*)(pool + OFF_OVL);   // Q_*3 floats
  float* tstage = stage + Q_ * 3;             // T_*3 floats
  const float* gout = outputs + (size_t)b * Q_ * 3;
  const float* gtgt = targets + (size_t)b * T_ * 3;
  for (int c = tid; c < (Q_ * 3) / 4; c += NTHREADS)   // 1536 x 16B
    async_copy16(gout + c * 4, stage + c * 4);
  for (int c = tid; c < (T_ * 3) / 4; c += NTHREADS)   // 192 x 16B
    async_copy16(gtgt + c * 4, tstage + c * 4);
  wait_async0();
  __syncthreads();

  // unpack SoA
  for (int q = tid; q < Q_; q += NTHREADS) {
    prob[q] = stage[q * 3 + 0];   // logits for now
    opx[q]  = stage[q * 3 + 1];
    opy[q]  = stage[q * 3 + 2];
  }
  for (int t = tid; t < T_; t += NTHREADS) {
    tpx[t] = tstage[t * 3 + 1];
    tpy[t] = tstage[t * 3 + 2];
  }
  __syncthreads();

  // ---------- Phase B: fp32 softmax over Q of the class logit ----------
  {
    float lm = -FLT_MAX;
    for (int q = tid; q < Q_; q += NTHREADS) lm = fmaxf(lm, prob[q]);
#pragma unroll
    for (int off = 16; off > 0; off >>= 1) lm = fmaxf(lm, __shfl_xor(lm, off, 32));
    if (lane == 0) rvalD[wid] = (double)lm;
    __syncthreads();
    if (wid == 0) {
      float x = (float)rvalD[lane];
#pragma unroll
      for (int off = 16; off > 0; off >>= 1) x = fmaxf(x, __shfl_xor(x, off, 32));
      if (lane == 0) sc->f0 = x;
    }
    __syncthreads();
    const float mx = sc->f0;
    __syncthreads();

    float ls = 0.f;
    for (int q = tid; q < Q_; q += NTHREADS) ls += expf(prob[q] - mx);
#pragma unroll
    for (int off = 16; off > 0; off >>= 1) ls += __shfl_xor(ls, off, 32);
    if (lane == 0) rvalD[wid] = (double)ls;
    __syncthreads();
    if (wid == 0) {
      float x = (float)rvalD[lane];
#pragma unroll
      for (int off = 16; off > 0; off >>= 1) x += __shfl_xor(x, off, 32);
      if (lane == 0) sc->f0 = x;
    }
    __syncthreads();
    const float invS = 1.0f / sc->f0;
    for (int q = tid; q < Q_; q += NTHREADS) prob[q] = expf(prob[q] - mx) * invS;
  }
  __syncthreads();   // staging dead from here; solver arrays may now reuse it

  // ---------- Phase C: Jonker-Volgenant on cost^T (n=T_ rows, m=Q_ cols) ----------
  for (int j = tid; j <= Q_; j += NTHREADS) { v[j] = 0.0; p[j] = 0; way[j] = 0; }
  for (int k = tid; k <= T_; k += NTHREADS) u[k] = 0.0;
  __syncthreads();

  for (int i = 1; i <= T_; ++i) {
    for (int j = tid; j <= Q_; j += NTHREADS) { minv[j] = (double)INFINITY; used[j] = 0; }
    if (tid == 0) { p[0] = i; sc->j0 = 0; used[0] = 1; }
    __syncthreads();

    while (true) {
      // ---- scan: relax minv with cur = cost(i0-1, j-1) - u[i0] - v[j] ----
      const int    j0  = sc->j0;
      const int    i0  = p[j0];
      const double ui0 = u[i0];
      const float  tx  = tpx[i0 - 1];
      const float  ty  = tpy[i0 - 1];

      double lval = (double)INFINITY;
      int    lidx = INT_MAX;
      for (int j = tid; j <= Q_; j += NTHREADS) {
        if (j >= 1 && !used[j]) {
          const int q = j - 1;
          // fp32 cost (matches jax), promoted to fp64 for the solve (matches numpy)
          const float  cc  = -prob[q] + fabsf(opx[q] - tx) + fabsf(opy[q] - ty);
          const double cur = (double)cc - ui0 - v[j];
          double mv = minv[j];
          if (cur < mv) { mv = cur; minv[j] = cur; way[j] = j0; }
          if (mv < lval || (mv == lval && j < lidx)) { lval = mv; lidx = j; }
        }
      }
      wave_min_pair(lval, lidx);
      if (lane == 0) { rvalD[wid] = lval; ridx[wid] = lidx; }
      __syncthreads();
      if (wid == 0) {
        double x = rvalD[lane]; int xi = ridx[lane];
        wave_min_pair(x, xi);
        if (lane == 0) { sc->delta = x; sc->j1 = xi; }
      }
      __syncthreads();

      // ---- apply delta; owner of j1 marks it used for the next scan ----
      const double delta = sc->delta;
      const int    j1    = sc->j1;
      for (int j = tid; j <= Q_; j += NTHREADS) {
        if (used[j]) { u[p[j]] += delta; v[j] -= delta; }  // p[j] distinct over used j
        else if (j >= 1) {
          minv[j] -= delta;                 // j1 is still unused here (as in reference)
          if (j == j1) used[j] = 1;         // becomes used for the next iteration
        }
      }
      if (tid == 0) { sc->j0 = j1; sc->done = (p[j1] == 0); }
      __syncthreads();
      if (sc->done) break;
    }

    if (tid == 0) {  // augment along alternating path
      int j0 = sc->j0;
      while (j0) { const int j1 = way[j0]; p[j0] = p[j1]; j0 = j1; }
    }
    __syncthreads();
  }

  // ---------- Phase D: extract assignment, stable argsort by rank ----------
  for (int j = tid; j <= Q_; j += NTHREADS)
    if (j >= 1 && p[j] > 0) ansA[p[j] - 1] = j - 1;
  __syncthreads();

  if (tid < T_) {
    const int a = ansA[tid];
    int r = 0;
    for (int t2 = 0; t2 < T_; ++t2) r += (ansA[t2] < a);  // distinct -> unique ranks
    row_out[b * T_ + r] = a;     // rows = ans[order]
    col_out[b * T_ + r] = tid;   // cols = order
  }
}

extern "C" void kernel_launch(void* const* d_in, const int* in_sizes, int n_in,
                              void* d_out, int out_size, void* d_ws, size_t ws_size,
                              hipStream_t stream) {
  (void)in_sizes; (void)n_in; (void)d_ws; (void)ws_size; (void)out_size;
  const float* outputs = (const float*)d_in[0];   // (8, 2048, 3) f32
  const float* targets = (const float*)d_in[1];   // (8, 256, 3)  f32
  // Reference returns (row_ind, col_ind), both (8,256) int32, concatenated flat.
  int* row_out = (int*)d_out;
  int* col_out = row_out + B_ * T_;
  hipLaunchKernelGGL(hungarian_kernel, dim3(B_), dim3(NTHREADS), 0, stream,
                     outputs, targets, row_out, col_out);
}